// LogLinearAttention_45483703664962
// MI455X (gfx1250) — compile-verified
//
#include <hip/hip_runtime.h>
#include <math.h>

#define B_ 8
#define S_ 2048
#define D_ 512

typedef float v2f __attribute__((ext_vector_type(2)));
typedef float v8f __attribute__((ext_vector_type(8)));

// ---------------------------------------------------------------------------
// K0: zero the padded xsum buffer (16 x 512 f32). Must run every launch since
// K1 accumulates with atomics and the harness does not re-poison between runs.
// ---------------------------------------------------------------------------
__global__ void k_zero(float* __restrict__ p, int n) {
  int i = blockIdx.x * blockDim.x + threadIdx.x;
  if (i < n) p[i] = 0.0f;
}

// ---------------------------------------------------------------------------
// K1: xsum[b][d] = sum_s x[b][s][d].  Streams all of x (33.5 MB) once.
// Block = 128 threads, each owns 4 consecutive d (float4 -> global_load_b128,
// fully coalesced 2KB rows). Grid = (S/32 chunks, B). Partial sums merged with
// global_atomic_add_f32 (64 adders per address -> trivial L2 contention).
// ---------------------------------------------------------------------------
__global__ void k_colsum(const float* __restrict__ x, float* __restrict__ xsum) {
  const int b  = blockIdx.y;
  const int s0 = blockIdx.x * 32;
  const int d4 = threadIdx.x * 4;                    // 128*4 = 512 columns
  const float* xp = x + ((size_t)b * S_ + s0) * D_ + d4;
  float4 acc = make_float4(0.f, 0.f, 0.f, 0.f);
#pragma unroll 4
  for (int s = 0; s < 32; ++s) {
    float4 v = *(const float4*)(xp + (size_t)s * D_);
    acc.x += v.x; acc.y += v.y; acc.z += v.z; acc.w += v.w;
  }
  float* o = xsum + b * D_ + d4;
  atomicAdd(o + 0, acc.x);
  atomicAdd(o + 1, acc.y);
  atomicAdd(o + 2, acc.z);
  atomicAdd(o + 3, acc.w);
}

// ---------------------------------------------------------------------------
// K2: pooled[m][n] = sum_k xsum[m][k] * Wr[n][k]   (bias added in K3)
// One wave (32 threads) per 16-wide N tile; 128 x V_WMMA_F32_16X16X4_F32.
// A-fragment layout (ISA 7.12.2, 32-bit 16x4):
//   lanes 0-15 : M = lane,     VGPR{0,1} = K{0,1}
//   lanes16-31 : M = lane-16,  VGPR{0,1} = K{2,3}
// B-fragment mirrors this over N; Wr rows are contiguous in k, so both
// fragments are single aligned float2 loads.
// ---------------------------------------------------------------------------
__global__ void k_gemm_wmma(const float* __restrict__ xsum,   // [16][512], rows 8..15 zero
                            const float* __restrict__ Wr,     // [512][512]
                            float* __restrict__ pooled) {     // [16][512]
  const int lane = threadIdx.x;        // 0..31, one full wave, EXEC all ones
  const int half = lane >> 4;          // K sub-block selector
  const int l15  = lane & 15;          // M index (A) / N index (B)
  const int n0   = blockIdx.x * 16;

  const float* arow = xsum + (size_t)l15 * D_ + 2 * half;
  const float* brow = Wr + (size_t)(n0 + l15) * D_ + 2 * half;

  v8f c = {};
#pragma unroll 8
  for (int k0 = 0; k0 < D_; k0 += 4) {
    v2f a = *(const v2f*)(arow + k0);
    v2f b = *(const v2f*)(brow + k0);
    // 8 args: (neg_a, A, neg_b, B, c_mod, C, reuse_a, reuse_b)
    c = __builtin_amdgcn_wmma_f32_16x16x4_f32(false, a, false, b,
                                              (short)0, c, false, false);
  }

  // D layout: VGPR j -> M=j (lanes 0-15), M=j+8 (lanes 16-31); N = lane&15
#pragma unroll
  for (int j = 0; j < 8; ++j) {
    const int m = j + 8 * half;
    pooled[(size_t)m * D_ + n0 + l15] = c[j];
  }
}

// ---------------------------------------------------------------------------
// K3: out[b] = sigmoid( sum_n (pooled[b,n] + S*br[n]) * Wl[n] + bl )
// ---------------------------------------------------------------------------
__global__ void k_head(const float* __restrict__ pooled,
                       const float* __restrict__ br,
                       const float* __restrict__ Wl,
                       const float* __restrict__ bl,
                       float* __restrict__ out) {
  __shared__ float red[256];
  const int b = blockIdx.x;
  const int t = threadIdx.x;
  float s = 0.f;
  for (int n = t; n < D_; n += 256)
    s += (pooled[(size_t)b * D_ + n] + (float)S_ * br[n]) * Wl[n];
  red[t] = s;
  __syncthreads();
  for (int w = 128; w > 0; w >>= 1) {
    if (t < w) red[t] += red[t + w];
    __syncthreads();
  }
  if (t == 0) {
    float z = red[0] + bl[0];
    out[b] = 1.0f / (1.0f + expf(-z));
  }
}

// ---------------------------------------------------------------------------
// inputs: 0:x 1:Wq 2:bq 3:Wv 4:bv 5:Wr 6:br 7:Wl 8:bl
// Wq/bq/Wv/bv provably do not affect the output (softmax columns sum to 1).
// ---------------------------------------------------------------------------
extern "C" void kernel_launch(void* const* d_in, const int* in_sizes, int n_in,
                              void* d_out, int out_size, void* d_ws, size_t ws_size,
                              hipStream_t stream) {
  (void)in_sizes; (void)n_in; (void)out_size; (void)ws_size;
  const float* x  = (const float*)d_in[0];
  const float* Wr = (const float*)d_in[5];
  const float* br = (const float*)d_in[6];
  const float* Wl = (const float*)d_in[7];
  const float* bl = (const float*)d_in[8];
  float* out = (float*)d_out;

  float* xsum   = (float*)d_ws;                                  // 16*512 f32
  float* pooled = (float*)((char*)d_ws + 16 * D_ * sizeof(float)); // 16*512 f32

  k_zero<<<(16 * D_) / 256, 256, 0, stream>>>(xsum, 16 * D_);

  dim3 g1(S_ / 32, B_);
  k_colsum<<<g1, 128, 0, stream>>>(x, xsum);

  k_gemm_wmma<<<D_ / 16, 32, 0, stream>>>(xsum, Wr, pooled);

  k_head<<<B_, 256, 0, stream>>>(pooled, br, Wl, bl, out);
}